// AssisExc_4930622456189
// MI455X (gfx1250) — compile-verified
//
#include <hip/hip_runtime.h>
#include <math.h>

typedef __attribute__((ext_vector_type(2))) float v2f;
typedef __attribute__((ext_vector_type(4))) float v4f;
typedef __attribute__((ext_vector_type(8))) float v8f;

#define N_    8
#define C_    256
#define H_    128
#define W_    128
#define G_    32
#define HW_   (H_ * W_)
#define TILE  256   // pixels per block (2 rows of W=128)

// One block = 256 threads = 8 wave32 = one 256-pixel tile of one image.
// Phase A: per-pixel box-union mask.
// Phase B: channel sum via V_WMMA_F32_16X16X4_F32 (A = 16px x 4ch of x, B = ones).
// Phase C: out = x + alpha/C * sum * mask, streamed with b128 loads/stores
//          (tile re-read is L2-resident; HBM traffic stays at the 268MB floor).
__global__ __launch_bounds__(256) void AssisExc_kernel(
    const float* __restrict__ x, const float* __restrict__ gt,
    const int* __restrict__ stride_p, const int* __restrict__ epoch_p,
    float* __restrict__ out)
{
    __shared__ __align__(16) float lds_sum[TILE];
    __shared__ __align__(16) float lds_exc[TILE];

    const int t    = threadIdx.x;
    const int lane = t & 31;
    const int wv   = t >> 5;
    const int blk  = blockIdx.x;
    const int n        = blk >> 6;            // 64 tiles per image
    const int tile_base = (blk & 63) * TILE;  // pixel offset in HxW plane

    const float inv_stride = 1.0f / (float)stride_p[0];
    const float alpha = 0.5f * (1.0f + cosf((float)M_PI * (float)epoch_p[0] / 22.0f));
    const float scale = alpha / (float)C_;

    // ---------------- Phase A: union-of-boxes mask for pixel `t` of the tile ----
    const int   q  = tile_base + t;
    const float wf = (float)(q & (W_ - 1));
    const float hf = (float)(q >> 7);
    const float* box = gt + (long)n * G_ * 4;
    bool inside = false;
    for (int g = 0; g < G_; ++g) {
        float x1 = floorf(box[g * 4 + 0] * inv_stride);
        float y1 = floorf(box[g * 4 + 1] * inv_stride);
        float x2 = ceilf (box[g * 4 + 2] * inv_stride);
        float y2 = ceilf (box[g * 4 + 3] * inv_stride);
        inside = inside || ((wf >= x1) && (wf < x2) && (hf >= y1) && (hf < y2));
    }

    // ---------------- Phase B: channel-sum via WMMA f32 16x16x4 -----------------
    // Wave `wv` owns pixels [wv*32, wv*32+32). Two accumulator chains of 16 px.
    // 32-bit A layout (16x4 MxK): lanes 0-15 -> M=lane, VGPR0=K0, VGPR1=K1;
    //                             lanes 16-31 -> M=lane-16, VGPR0=K2, VGPR1=K3.
    const long img_base = (long)n * C_ * HW_;
    const int  base_p   = wv * 32;
    const int  m        = lane & 15;
    const int  chan_off = (lane >> 4) << 1;   // 0 or 2
    const float* p0 = x + img_base + tile_base + base_p + m;       // pixels base_p..+15
    const float* p1 = p0 + 16;                                     // pixels base_p+16..+31

    v8f acc0 = {};
    v8f acc1 = {};
    v2f bones; bones.x = 1.0f; bones.y = 1.0f;   // B = 4x16 all-ones

    for (int k = 0; k < 64; ++k) {
        const long c = (long)(4 * k + chan_off);
        v2f a0, a1;
        a0.x = p0[c * HW_];
        a0.y = p0[(c + 1) * HW_];
        a1.x = p1[c * HW_];
        a1.y = p1[(c + 1) * HW_];
        __builtin_prefetch(p0 + (c + 8) * (long)HW_, 0, 0);   // global_prefetch_b8
        acc0 = __builtin_amdgcn_wmma_f32_16x16x4_f32(
                   false, a0, false, bones, (short)0, acc0, false, false);
        acc1 = __builtin_amdgcn_wmma_f32_16x16x4_f32(
                   false, a1, false, bones, (short)0, acc1, false, false);
    }

    // D layout (16x16 f32): VGPR r, lanes 0-15 -> M=r, lanes 16-31 -> M=8+r;
    // all 16 columns equal the channel sum, so lanes 0 and 16 extract rows.
    if ((lane & 15) == 0) {
        const int half = lane >> 4;  // 0 or 1
        #pragma unroll
        for (int r = 0; r < 8; ++r) {
            lds_sum[base_p + 8 * half + r]      = acc0[r];
            lds_sum[base_p + 16 + 8 * half + r] = acc1[r];
        }
    }
    __syncthreads();

    lds_exc[t] = inside ? lds_sum[t] * scale : 0.0f;
    __syncthreads();

    // ---------------- Phase C: out = x + exc, b128 streaming --------------------
    const float* xin  = x   + img_base + tile_base;
    float*       xout = out + img_base + tile_base;
    #pragma unroll 4
    for (int i = 0; i < 64; ++i) {
        const int flat = (i * 256 + t) * 4;   // index into [C=256][P=256] tile
        const int c    = flat >> 8;           // channel
        const int p    = flat & 255;          // pixel in tile (multiple of 4)
        const long off = (long)c * HW_ + p;
        v4f xv = *(const v4f*)(xin + off);    // global_load_b128, 512B/wave
        v4f ev = *(const v4f*)&lds_exc[p];    // ds_load_b128
        xv.x += ev.x; xv.y += ev.y; xv.z += ev.z; xv.w += ev.w;
        *(v4f*)(xout + off) = xv;             // global_store_b128
    }
}

extern "C" void kernel_launch(void* const* d_in, const int* in_sizes, int n_in,
                              void* d_out, int out_size, void* d_ws, size_t ws_size,
                              hipStream_t stream) {
    const float* x        = (const float*)d_in[0];
    const float* gt       = (const float*)d_in[1];
    const int*   stride_p = (const int*)  d_in[2];
    const int*   epoch_p  = (const int*)  d_in[3];
    float*       out      = (float*)d_out;

    dim3 grid(N_ * H_ * W_ / TILE);  // 512 blocks
    dim3 block(256);                 // 8 waves -> fills a WGP twice
    hipLaunchKernelGGL(AssisExc_kernel, grid, block, 0, stream,
                       x, gt, stride_p, epoch_p, out);
}